// LearnabilityMB_38878043963739
// MI455X (gfx1250) — compile-verified
//
#include <hip/hip_runtime.h>
#include <hip/hip_bf16.h>

typedef __attribute__((ext_vector_type(2))) float v2f;
typedef __attribute__((ext_vector_type(8))) float v8f;

#define NROWS 8192
#define NCOLS 4096
#define SEGS 16
#define ROWS_PER_SEG (NROWS / SEGS)       /* 512 */
#define TILES_PER_SEG (ROWS_PER_SEG / 16) /* 32 */
#define LDS_STRIDE 20                     /* floats; 80B rows: 16B-aligned, conflict-free */

// ---------------------------------------------------------------------------
// Pass A: per-segment column sums of w = X * v   (full-bandwidth streaming)
// SC[seg][d] = sum_{j in seg} X[j,d] * v[j]
// ---------------------------------------------------------------------------
__global__ __launch_bounds__(256)
void phase_colsum(const float* __restrict__ X, const float* __restrict__ vin,
                  float* __restrict__ SC)
{
    const int d   = blockIdx.x * 256 + threadIdx.x;
    const int seg = blockIdx.y;
    const int r0  = seg * ROWS_PER_SEG;
    float a0 = 0.f, a1 = 0.f, a2 = 0.f, a3 = 0.f;
    for (int j = 0; j < ROWS_PER_SEG; j += 4) {
        const int r = r0 + j;
        a0 += X[(size_t)(r + 0) * NCOLS + d] * vin[r + 0];
        a1 += X[(size_t)(r + 1) * NCOLS + d] * vin[r + 1];
        a2 += X[(size_t)(r + 2) * NCOLS + d] * vin[r + 2];
        a3 += X[(size_t)(r + 3) * NCOLS + d] * vin[r + 3];
    }
    SC[(size_t)seg * NCOLS + d] = (a0 + a1) + (a2 + a3);
}

// ---------------------------------------------------------------------------
// Pass B prep: exclusive scan of segment sums along the scan direction.
// suffix=1: CR[q][d] = sum_{p>q} SC[p][d];  suffix=0: CR[q][d] = sum_{p<q}.
// ---------------------------------------------------------------------------
__global__ __launch_bounds__(256)
void phase_carry(const float* __restrict__ SC, float* __restrict__ CR, int suffix)
{
    const int d = blockIdx.x * 256 + threadIdx.x;
    float acc = 0.f;
    if (suffix) {
        for (int q = SEGS - 1; q >= 0; --q) {
            CR[(size_t)q * NCOLS + d] = acc;
            acc += SC[(size_t)q * NCOLS + d];
        }
    } else {
        for (int q = 0; q < SEGS; ++q) {
            CR[(size_t)q * NCOLS + d] = acc;
            acc += SC[(size_t)q * NCOLS + d];
        }
    }
}

// ---------------------------------------------------------------------------
// Async DMA of one 16x16 f32 tile (this wave's 16-column strip) into LDS.
// Two GLOBAL_LOAD_ASYNC_TO_LDS_B128: 16B per lane, rows 0-7 then 8-15.
// Tracked by ASYNCcnt; completes in order.
// ---------------------------------------------------------------------------
__device__ __forceinline__ void async_tile_load(const float* __restrict__ X,
                                                int rowbase, int wcb,
                                                float* ldsbuf, int lane)
{
    const int r  = lane >> 2;        // 0..7
    const int cg = (lane & 3) * 4;   // 0,4,8,12 (floats)
    const float* g0 = X + (size_t)(rowbase + r) * NCOLS + wcb + cg;
    const float* g1 = X + (size_t)(rowbase + 8 + r) * NCOLS + wcb + cg;
    const unsigned l0 = (unsigned)(uintptr_t)(ldsbuf + r * LDS_STRIDE + cg);
    const unsigned l1 = (unsigned)(uintptr_t)(ldsbuf + (8 + r) * LDS_STRIDE + cg);
    asm volatile("global_load_async_to_lds_b128 %0, %1, off"
                 :: "v"(l0), "v"((unsigned long long)(uintptr_t)g0) : "memory");
    asm volatile("global_load_async_to_lds_b128 %0, %1, off"
                 :: "v"(l1), "v"((unsigned long long)(uintptr_t)g1) : "memory");
}

// ---------------------------------------------------------------------------
// Pass B: intra-segment scan + emit.
// Each wave owns 16 columns; walks 32 tiles of 16 rows (direction depends on
// suffix/prefix). Tiles are double-buffered in LDS via async-to-LDS DMA.
// The strict-triangular intra-tile scan is T(16x16 ones) @ W(16x16) done as
// 4 chained V_WMMA_F32_16X16X4_F32 (K=16 split 4x4), all fp32.
// vout[i] += sum over this wave's columns of X[i,d]*s[d]*Sshift[i,d].
// ---------------------------------------------------------------------------
__global__ __launch_bounds__(128)
void phase_scan(const float* __restrict__ X, const float* __restrict__ s,
                const float* __restrict__ vin, const float* __restrict__ CR,
                float* __restrict__ vout, int suffix)
{
    __shared__ float lds[4][2][16][LDS_STRIDE];   // per-wave double buffer

    const int lane = threadIdx.x & 31;
    const int wv   = threadIdx.x >> 5;
    const int half = lane >> 4;        // 0: lanes 0-15, 1: lanes 16-31
    const int n    = lane & 15;        // column-within-wave (B/C) and M index (A)
    const int kb   = half * 2;         // K-pair base selected by lane half
    const int wcb  = blockIdx.x * 64 + wv * 16;   // wave column base
    const int col  = wcb + n;
    const int seg  = blockIdx.y;
    const int segbase = seg * ROWS_PER_SEG;

    const float sv = s[col];
    float acc = CR[(size_t)seg * NCOLS + col];   // carry-in from other segments

    // A = strict triangular ones, 16x4 f32 slabs. ISA layout:
    // lanes 0-15: M=lane, K={0,1}; lanes 16-31: M=lane-16, K={2,3}.
    v2f A[4];
#pragma unroll
    for (int q = 0; q < 4; ++q) {
        const int j0 = 4 * q + kb;
        if (suffix) {  // Sshift[i] = sum_{j>i} w[j]
            A[q].x = (j0     > n) ? 1.0f : 0.0f;
            A[q].y = (j0 + 1 > n) ? 1.0f : 0.0f;
        } else {       // Sshift[i] = sum_{j<i} w[j]
            A[q].x = (j0     < n) ? 1.0f : 0.0f;
            A[q].y = (j0 + 1 < n) ? 1.0f : 0.0f;
        }
    }

    // Preload first tile into buffer 0.
    {
        const int t0 = suffix ? (TILES_PER_SEG - 1) : 0;
        async_tile_load(X, segbase + t0 * 16, wcb, &lds[wv][0][0][0], lane);
    }

    for (int tt = 0; tt < TILES_PER_SEG; ++tt) {
        const int t = suffix ? (TILES_PER_SEG - 1 - tt) : tt;
        const int rowbase = segbase + t * 16;
        const int cur = tt & 1;

        if (tt + 1 < TILES_PER_SEG) {
            const int tn = suffix ? (t - 1) : (t + 1);
            // Buffer we are about to overwrite was last read in the previous iteration;
            // make sure those LDS reads have drained before the DMA lands.
            asm volatile("s_wait_dscnt 0x0" ::: "memory");
            async_tile_load(X, segbase + tn * 16, wcb, &lds[wv][cur ^ 1][0][0], lane);
            // Current tile's 2 loads complete before the 2 just issued.
            asm volatile("s_wait_asynccnt 0x2" ::: "memory");
        } else {
            asm volatile("s_wait_asynccnt 0x0" ::: "memory");
        }

        // v values for the tile's 16 rows, one per lane (lane&15 = row).
        const float vv = vin[rowbase + n];

        const float (*W)[LDS_STRIDE] = lds[wv][cur];

        // Build B = W rows (4x16 f32 slabs) from LDS; accumulate lane colsum.
        v2f B[4];
        float cs = 0.f;
#pragma unroll
        for (int q = 0; q < 4; ++q) {
            const int j0 = 4 * q + kb;
            const float v0 = __shfl(vv, j0, 32);
            const float v1 = __shfl(vv, j0 + 1, 32);
            const float w0 = W[j0][n] * v0;
            const float w1 = W[j0 + 1][n] * v1;
            B[q].x = w0; B[q].y = w1;
            cs += w0 + w1;
        }

        // Intra-tile strict scan: C = T @ W via 4 chained f32 WMMAs (K=16).
        v8f C = {};
#pragma unroll
        for (int q = 0; q < 4; ++q)
            C = __builtin_amdgcn_wmma_f32_16x16x4_f32(
                    false, A[q], false, B[q], (short)0, C, false, false);

        // Full per-column tile sum (halves covered disjoint row sets).
        cs += __shfl_xor(cs, 16, 32);

        // Emit: contrib[row m] = X[m,col] * s[col] * (Sshift_local + carry).
        float contrib[8];
#pragma unroll
        for (int k = 0; k < 8; ++k)
            contrib[k] = W[k + half * 8][n] * sv * (C[k] + acc);

        // Reduce across the 16 lanes of each half (columns of this wave).
#pragma unroll
        for (int k = 0; k < 8; ++k) {
            float v = contrib[k];
            v += __shfl_xor(v, 1, 32);
            v += __shfl_xor(v, 2, 32);
            v += __shfl_xor(v, 4, 32);
            v += __shfl_xor(v, 8, 32);
            contrib[k] = v;
        }
        if (n == 0) {  // lane 0 -> rows 0..7, lane 16 -> rows 8..15
#pragma unroll
            for (int k = 0; k < 8; ++k)
                atomicAdd(&vout[rowbase + k + half * 8], contrib[k]);
        }
        acc += cs;  // advance carry past this tile
    }
}

// ---------------------------------------------------------------------------
// Final: four dot products + scalar combine.  out = 1 - sum_i a[i]*t_i
// ---------------------------------------------------------------------------
__global__ __launch_bounds__(256)
void finalize(const float* __restrict__ y, const float* __restrict__ a,
              const float* __restrict__ r1, const float* __restrict__ l1,
              const float* __restrict__ r2, const float* __restrict__ l2,
              float* __restrict__ out)
{
    __shared__ float red[4][256];
    const int tid = threadIdx.x;
    float t1 = 0.f, t2 = 0.f, t3 = 0.f, t4 = 0.f;
    for (int i = tid; i < NROWS; i += 256) {
        const float yi = y[i], r1i = r1[i], l1i = l1[i], r2i = r2[i], l2i = l2[i];
        t1 += yi * r1i;   // a0 * (y  . r1)
        t2 += l1i * r1i;  // a1 * (l1 . r1)
        t3 += l1i * r2i;  // a2 * (l1 . r2)
        t4 += l2i * r2i;  // a3 * (l2 . r2)
    }
    red[0][tid] = t1; red[1][tid] = t2; red[2][tid] = t3; red[3][tid] = t4;
    __syncthreads();
    for (int sH = 128; sH > 0; sH >>= 1) {
        if (tid < sH) {
#pragma unroll
            for (int q = 0; q < 4; ++q) red[q][tid] += red[q][tid + sH];
        }
        __syncthreads();
    }
    if (tid == 0)
        out[0] = 1.0f - (a[0] * red[0][0] + a[1] * red[1][0] +
                         a[2] * red[2][0] + a[3] * red[3][0]);
}

// ---------------------------------------------------------------------------
extern "C" void kernel_launch(void* const* d_in, const int* in_sizes, int n_in,
                              void* d_out, int out_size, void* d_ws, size_t ws_size,
                              hipStream_t stream)
{
    const float* s = (const float*)d_in[0];   // [D,1]
    const float* X = (const float*)d_in[1];   // [N,D]
    const float* y = (const float*)d_in[2];   // [N,1]
    const float* a = (const float*)d_in[3];   // [4]
    float* out = (float*)d_out;

    float* ws = (float*)d_ws;
    float* r1 = ws;
    float* l1 = ws + NROWS;
    float* r2 = ws + 2 * NROWS;
    float* l2 = ws + 3 * NROWS;
    float* SC = ws + 4 * NROWS;                    // [SEGS][NCOLS]
    float* CR = SC + (size_t)SEGS * NCOLS;         // [SEGS][NCOLS]

    // Zero the four output vectors (accumulated via atomics).
    hipMemsetAsync(r1, 0, (size_t)4 * NROWS * sizeof(float), stream);

    struct Ph { const float* vin; float* vout; int suffix; };
    const Ph ph[4] = {
        { y,  r1, 1 },   // r1 = strict_upper(K) @ y
        { y,  l1, 0 },   // l1 = strict_lower(K) @ y
        { r1, r2, 1 },   // r2 = strict_upper(K) @ r1
        { l1, l2, 0 },   // l2 = strict_lower(K) @ l1
    };

    for (int p = 0; p < 4; ++p) {
        phase_colsum<<<dim3(NCOLS / 256, SEGS), 256, 0, stream>>>(X, ph[p].vin, SC);
        phase_carry <<<dim3(NCOLS / 256),       256, 0, stream>>>(SC, CR, ph[p].suffix);
        phase_scan  <<<dim3(NCOLS / 64,  SEGS), 128, 0, stream>>>(
            X, s, ph[p].vin, CR, ph[p].vout, ph[p].suffix);
    }

    finalize<<<1, 256, 0, stream>>>(y, a, r1, l1, r2, l2, out);
}